// SalientRegionSeparationLoss_16037407883702
// MI455X (gfx1250) — compile-verified
//
#include <hip/hip_runtime.h>
#include <hip/hip_bf16.h>
#include <math.h>

// ---------------------------------------------------------------------------
// SalientRegionSeparationLoss for MI455X (gfx1250, wave32, WMMA).
//
// D_ij = margin - p_i + q_j is computed as a rank-2 f32 WMMA:
//   A (16x4):  row i = [p_i, 1, 0, 0]
//   B (4x16):  col j = [-1, q_j + margin, 0, 0]^T     (margin pre-baked into Q)
//   D = A x B  (exact in f32: coefficients are +/-1 and 1)
// Sentinel padding (+1e30 for non-pos, -1e30 for non-neg) makes every
// invalid pair contribute relu(-huge) = 0, so no compaction is needed.
// relu is a single v_max_num_f32 via inline asm (skips clang's canonicalize;
// WMMA outputs here are always finite), accumulated into 8 independent
// chains that the compiler packs into v_pk_add_f32.
// ---------------------------------------------------------------------------

typedef __attribute__((ext_vector_type(2))) float v2f;
typedef __attribute__((ext_vector_type(8))) float v8f;

#define BATCH 8
#define NELEM 4096          // 64*64
#define MARGIN 0.5f
#define NTILES (NELEM / 16) // 256 tiles of 16

// Workspace layout (floats)
#define WS_P    0
#define WS_Q    (BATCH * NELEM)
#define WS_ACC  (2 * BATCH * NELEM)
#define WS_NPOS (WS_ACC + BATCH)
#define WS_NNEG (WS_NPOS + BATCH)

// Single-instruction relu: v_max_num_f32 dst, 0, src  (inline const in src0;
// VOP2 src1 must be a VGPR). Inputs are finite by construction, so skipping
// the IEEE canonicalize clang would otherwise insert is safe.
__device__ __forceinline__ float relu1(float x) {
    float r;
    asm("v_max_num_f32 %0, 0, %1" : "=v"(r) : "v"(x));
    return r;
}

// ---- Kernel 1: per-sample max + counts, build padded P/Q, zero accumulators
__global__ void srsl_prep_kernel(const float* __restrict__ x,
                                 const float* __restrict__ m,
                                 float* __restrict__ ws) {
    const int b   = blockIdx.x;     // one block per sample
    const int tid = threadIdx.x;    // 256 threads
    const float* xb = x + b * NELEM;
    const float* mb = m + b * NELEM;

    __shared__ float smax[256];
    __shared__ float spos[256];
    __shared__ float sneg[256];

    float lmax = -INFINITY, lp = 0.0f, ln = 0.0f;
    for (int i = tid; i < NELEM; i += 256) {
        float v  = xb[i];
        float mm = mb[i];
        lmax = fmaxf(lmax, v);
        lp  += (mm > 0.5f) ? 1.0f : 0.0f;
        ln  += (mm < 0.5f) ? 1.0f : 0.0f;
    }
    smax[tid] = lmax; spos[tid] = lp; sneg[tid] = ln;
    __syncthreads();
    for (int s = 128; s > 0; s >>= 1) {
        if (tid < s) {
            smax[tid] = fmaxf(smax[tid], smax[tid + s]);
            spos[tid] += spos[tid + s];
            sneg[tid] += sneg[tid + s];
        }
        __syncthreads();
    }
    const float mx = smax[0];
    if (tid == 0) {
        ws[WS_NPOS + b] = spos[0];
        ws[WS_NNEG + b] = sneg[0];
        ws[WS_ACC  + b] = 0.0f;     // zero pairwise accumulator
    }

    float* P = ws + WS_P + b * NELEM;
    float* Q = ws + WS_Q + b * NELEM;
    for (int i = tid; i < NELEM; i += 256) {
        float v  = xb[i] / mx;      // match reference: x / max(x)
        float mm = mb[i];
        P[i] = (mm > 0.5f) ? v :  1e30f;            // non-pos sentinel
        Q[i] = (mm < 0.5f) ? v + MARGIN : -1e30f;   // margin baked in; sentinel
    }
}

// ---- Kernel 2: tiled pairwise hinge via V_WMMA_F32_16X16X4_F32
// Grid: 32 blocks per sample * 8 samples = 256 blocks of 256 threads (8 waves).
// Each wave owns one 16-wide pos-tile (A loaded once, reused 256x) and sweeps
// all 256 neg-tiles. B-operand built branchlessly; relu = 1 v_max_num_f32;
// 8 independent accumulator chains -> v_pk_add_f32, co-issued with WMMA pipe.
__global__ void srsl_pairwise_wmma_kernel(const float* __restrict__ ws_ro,
                                          float* __restrict__ ws) {
    const int wave = threadIdx.x >> 5;
    const int lane = threadIdx.x & 31;
    const int b        = blockIdx.x >> 5;                  // sample
    const int pos_tile = ((blockIdx.x & 31) << 3) + wave;  // 0..255

    const float* P  = ws_ro + WS_P + b * NELEM + pos_tile * 16;
    const float* Qm = ws_ro + WS_Q + b * NELEM;            // q + margin

    const bool lo  = (lane < 16);
    const int  l15 = lane & 15;

    // A (16x4 f32): lanes 0-15 hold row M=lane, K=0..1; lanes 16-31 (K=2..3) zero
    const float pv = P[l15];            // full-wave load (halves share cacheline)
    v2f a;
    a.x = lo ? pv   : 0.0f;
    a.y = lo ? 1.0f : 0.0f;
    const float bx = lo ? -1.0f : 0.0f;

    float acc0 = 0.0f, acc1 = 0.0f, acc2 = 0.0f, acc3 = 0.0f;
    float acc4 = 0.0f, acc5 = 0.0f, acc6 = 0.0f, acc7 = 0.0f;

#pragma unroll 4
    for (int nt = 0; nt < NTILES; ++nt) {
        const float q = Qm[nt * 16 + l15];   // full-wave load, no divergence
        v2f bv;
        bv.x = bx;
        bv.y = lo ? q : 0.0f;

        v8f c = {};
        // D = A x B + 0  ->  D_ij = -p_i + (q_j + margin)   (exact in f32)
        v8f d = __builtin_amdgcn_wmma_f32_16x16x4_f32(
            /*neg_a=*/false, a, /*neg_b=*/false, bv,
            /*c_mod=*/(short)0, c, /*reuse_a=*/false, /*reuse_b=*/false);

        // relu: single v_max_num_f32 each; independent chains for packing
        acc0 += relu1(d[0]);
        acc1 += relu1(d[1]);
        acc2 += relu1(d[2]);
        acc3 += relu1(d[3]);
        acc4 += relu1(d[4]);
        acc5 += relu1(d[5]);
        acc6 += relu1(d[6]);
        acc7 += relu1(d[7]);
    }

    float sum = ((acc0 + acc1) + (acc2 + acc3)) + ((acc4 + acc5) + (acc6 + acc7));

    // wave32 reduction of per-lane partials
    for (int off = 16; off > 0; off >>= 1)
        sum += __shfl_xor(sum, off, 32);

    if (lane == 0)
        atomicAdd(&ws[WS_ACC + b], sum);
}

// ---- Kernel 3: normalize per sample, validity check, batch mean
__global__ void srsl_final_kernel(const float* __restrict__ ws,
                                  float* __restrict__ out) {
    if (threadIdx.x == 0 && blockIdx.x == 0) {
        float total = 0.0f;
        for (int b = 0; b < BATCH; ++b) {
            float np = ws[WS_NPOS + b];
            float nn = ws[WS_NNEG + b];
            if (np > 0.0f && nn > 0.0f)
                total += ws[WS_ACC + b] / (np * nn + 1e-8f);
        }
        out[0] = total / (float)BATCH;
    }
}

extern "C" void kernel_launch(void* const* d_in, const int* in_sizes, int n_in,
                              void* d_out, int out_size, void* d_ws, size_t ws_size,
                              hipStream_t stream) {
    const float* pred = (const float*)d_in[0];   // [8, 64, 64] f32
    const float* mask = (const float*)d_in[1];   // [8, 64, 64] f32
    float* out = (float*)d_out;                  // scalar f32
    float* ws  = (float*)d_ws;                   // needs ~262 KB

    srsl_prep_kernel<<<BATCH, 256, 0, stream>>>(pred, mask, ws);
    srsl_pairwise_wmma_kernel<<<BATCH * 32, 256, 0, stream>>>(ws, ws);
    srsl_final_kernel<<<1, 32, 0, stream>>>(ws, out);
}